// AudioPreprocessing_44598940402142
// MI455X (gfx1250) — compile-verified
//
#include <hip/hip_runtime.h>
#include <math.h>

// ---------------- problem constants ----------------
#define NW       5292000   // waveform samples
#define HOPSZ    196
#define NFFT     1024
#define PADL     512
#define NFRAMES  27001     // 1 + NW/HOP
#define MTILES   1688      // ceil(27001/16) -> 27008 rows
#define KFP      528       // 513 freq bins padded to 33*16
#define NTILE_F  33
#define NMELS    128
#define TW       27008     // padded time stride for mel/blur images

typedef float v2f __attribute__((ext_vector_type(2)));
typedef float v8f __attribute__((ext_vector_type(8)));

// ---------------- ordered-float helpers (radix select on bits) ------------
__device__ __forceinline__ unsigned f2o(float f) {
    unsigned u = __float_as_uint(f);
    return (u & 0x80000000u) ? ~u : (u | 0x80000000u);
}
__device__ __forceinline__ float o2f(unsigned u) {
    u = (u & 0x80000000u) ? (u & 0x7FFFFFFFu) : ~u;
    return __uint_as_float(u);
}

// ---------------- K0a: windowed DFT tables (transposed: [freq][n]) --------
__global__ void ap_init_dft(float* __restrict__ ccosT, float* __restrict__ csinT) {
    int idx = blockIdx.x * 256 + threadIdx.x;
    if (idx >= KFP * NFFT) return;
    int kf = idx >> 10;
    int n  = idx & 1023;
    const float C = 6.28318530717958647692f / 1024.0f;
    float w = 0.5f * (1.0f - cosf((float)n * C));      // Hann
    int r = (kf * n) & 1023;                            // exact arg reduction
    float ang = (float)r * C;
    ccosT[idx] = w * cosf(ang);
    csinT[idx] = w * sinf(ang);
}

// ---------------- K0b: mel filterbank (transposed: [mel][freq]) -----------
__global__ void ap_init_fb(float* __restrict__ fbT) {
    int idx = blockIdx.x * 256 + threadIdx.x;
    if (idx >= NMELS * KFP) return;
    int m = idx / KFP;
    int k = idx - m * KFP;
    if (k >= 513) { fbT[idx] = 0.0f; return; }
    float mel_lo = 2595.0f * log10f(1.0f + 500.0f   / 700.0f);
    float mel_hi = 2595.0f * log10f(1.0f + 15000.0f / 700.0f);
    float step = (mel_hi - mel_lo) / 129.0f;
    float f0 = 700.0f * (exp10f((mel_lo + step * (float)(m    )) / 2595.0f) - 1.0f);
    float f1 = 700.0f * (exp10f((mel_lo + step * (float)(m + 1)) / 2595.0f) - 1.0f);
    float f2 = 700.0f * (exp10f((mel_lo + step * (float)(m + 2)) / 2595.0f) - 1.0f);
    float freq = (float)k * (22050.0f / 512.0f);
    float down = (freq - f0) / (f1 - f0);
    float up   = (f2 - freq) / (f2 - f1);
    fbT[idx] = fmaxf(0.0f, fminf(down, up));
}

// ---------------- K1: power spectrum via f32 WMMA GEMMs -------------------
// spec[t][kf] = (frames@cosT)^2 + (frames@sinT)^2 ; 8 waves share one LDS
// frame panel (16 frames x 1024, XOR-swizzled for bank-conflict-free A-frags)
__global__ __launch_bounds__(256) void ap_spec(const float* __restrict__ wave,
                                               const float* __restrict__ ccosT,
                                               const float* __restrict__ csinT,
                                               float* __restrict__ spec) {
    __shared__ float As[16 * 1024];   // 64 KB
    const int tid   = threadIdx.x;
    const int tileM = blockIdx.x;

    for (int idx = tid; idx < 16 * 1024; idx += 256) {
        int row = idx >> 10;
        int n   = idx & 1023;
        int g = (tileM * 16 + row) * HOPSZ + n - PADL;   // reflect pad
        g = (g < 0) ? -g : g;
        g = (g >= NW) ? (2 * NW - 2 - g) : g;
        As[(row << 10) | (n ^ (row << 2))] = wave[g];    // XOR-4 bank swizzle
    }
    __syncthreads();

    int wv    = tid >> 5;
    int ntile = blockIdx.y * 8 + wv;
    if (ntile >= NTILE_F) return;                        // wave-uniform exit
    int lane = tid & 31;
    int half = lane >> 4;
    int lrow = lane & 15;
    int n0   = ntile << 4;
    const float* bC = ccosT + (size_t)(n0 + lrow) * NFFT;
    const float* bS = csinT + (size_t)(n0 + lrow) * NFFT;
    __builtin_prefetch(bC, 0, 0);
    __builtin_prefetch(bS, 0, 0);
    const int r4 = lrow << 2;

    v8f accC = {0.f,0.f,0.f,0.f,0.f,0.f,0.f,0.f};
    v8f accS = {0.f,0.f,0.f,0.f,0.f,0.f,0.f,0.f};
    #pragma unroll 4
    for (int kk = 0; kk < 256; ++kk) {
        int k0 = kk << 2;
        int kb = k0 + half * 2;
        v2f a  = *(const v2f*)&As[(lrow << 10) + (k0 ^ r4) + half * 2];
        v2f bc = *(const v2f*)&bC[kb];
        v2f bs = *(const v2f*)&bS[kb];
        accC = __builtin_amdgcn_wmma_f32_16x16x4_f32(false, a, false, bc,
                                                     (short)0, accC, false, false);
        accS = __builtin_amdgcn_wmma_f32_16x16x4_f32(false, a, false, bs,
                                                     (short)0, accS, false, false);
    }
    int t0  = tileM * 16 + half * 8;
    int col = n0 + lrow;
    #pragma unroll
    for (int r = 0; r < 8; ++r) {
        float c = accC[r], s = accS[r];
        spec[(size_t)(t0 + r) * KFP + col] = c * c + s * s;
    }
}

// ---------------- K2: mel GEMM via f32 WMMA, store transposed -------------
// mel[mel][t] = sum_k spec[t][k] * fb[k][mel]
__global__ __launch_bounds__(256) void ap_mel(const float* __restrict__ spec,
                                              const float* __restrict__ fbT,
                                              float* __restrict__ mel) {
    const int tid   = threadIdx.x;
    const int tileM = blockIdx.x;
    int wv   = tid >> 5;          // 8 waves -> 8 mel tiles (128 mels)
    int lane = tid & 31;
    int half = lane >> 4;
    int lrow = lane & 15;
    int n0   = wv << 4;
    const float* aBase = spec + (size_t)(tileM * 16 + lrow) * KFP;
    const float* bBase = fbT  + (size_t)(n0 + lrow) * KFP;

    v8f acc = {0.f,0.f,0.f,0.f,0.f,0.f,0.f,0.f};
    #pragma unroll 4
    for (int kk = 0; kk < KFP / 4; ++kk) {
        int kb = (kk << 2) + half * 2;
        v2f a = *(const v2f*)&aBase[kb];
        v2f b = *(const v2f*)&bBase[kb];
        acc = __builtin_amdgcn_wmma_f32_16x16x4_f32(false, a, false, b,
                                                    (short)0, acc, false, false);
    }
    int t0 = tileM * 16 + half * 8;
    int mi = n0 + lrow;
    #pragma unroll
    for (int r = 0; r < 8; ++r) {
        int t = t0 + r;
        if (t < NFRAMES) mel[(size_t)mi * TW + t] = acc[r];
    }
}

// ---------------- K3: 5x5 median blur (edge clamp) ------------------------
// Normalization elided: median & quantile thresholds commute with the
// strictly-increasing affine normalization, so the final mask is identical.
__global__ __launch_bounds__(256) void ap_blur(const float* __restrict__ mel,
                                               float* __restrict__ blur) {
    int t = blockIdx.x * 256 + threadIdx.x;
    int m = blockIdx.y;
    if (t >= NFRAMES) return;
    float v[25];
    #pragma unroll
    for (int dm = -2; dm <= 2; ++dm) {
        int mm = min(max(m + dm, 0), NMELS - 1);
        #pragma unroll
        for (int dt = -2; dt <= 2; ++dt) {
            int tt = min(max(t + dt, 0), NFRAMES - 1);
            v[(dm + 2) * 5 + (dt + 2)] = mel[(size_t)mm * TW + tt];
        }
    }
    #pragma unroll
    for (int i = 0; i <= 12; ++i) {
        #pragma unroll
        for (int j = i + 1; j < 25; ++j) {
            float a = v[i], b = v[j];
            v[i] = fminf(a, b);
            v[j] = fmaxf(a, b);
        }
    }
    blur[(size_t)m * TW + t] = v[12];
}

// ---------------- K4: per-row 75th percentile (exact order stat 20250) ----
__global__ __launch_bounds__(256) void ap_rowthr(const float* __restrict__ blur,
                                                 float* __restrict__ thrRow) {
    __shared__ unsigned hist[256];
    __shared__ unsigned sel[2];
    int row = blockIdx.x, tid = threadIdx.x;
    unsigned prefix = 0;
    int k = 20250;                               // 0.75*(27001-1) exactly
    for (int pass = 0; pass < 4; ++pass) {
        int shift = 24 - pass * 8;
        unsigned hmask = (pass == 0) ? 0u : (0xFFFFFFFFu << (shift + 8));
        hist[tid] = 0;
        __syncthreads();
        for (int i = tid; i < NFRAMES; i += 256) {
            unsigned u = f2o(blur[(size_t)row * TW + i]);
            if ((u & hmask) == prefix) atomicAdd(&hist[(u >> shift) & 255], 1u);
        }
        __syncthreads();
        if (tid == 0) {
            unsigned cum = 0, b = 0;
            for (; b < 256; ++b) {
                unsigned h = hist[b];
                if (cum + h > (unsigned)k) break;
                cum += h;
            }
            sel[0] = b;
            sel[1] = (unsigned)k - cum;
        }
        __syncthreads();
        prefix |= sel[0] << shift;
        k = (int)sel[1];
        __syncthreads();
    }
    if (tid == 0) thrRow[row] = o2f(prefix);
}

// ---------------- K5: per-column quantile (idx 95.25 -> interp 95th/96th) --
__device__ unsigned ap_radix_select(const unsigned* col, int stride, int n, int k) {
    unsigned prefix = 0;
    for (int bit = 31; bit >= 0; --bit) {
        unsigned hmask = (bit == 31) ? 0u : (0xFFFFFFFFu << (bit + 1));
        int cnt0 = 0;
        for (int i = 0; i < n; ++i) {
            unsigned u = col[i * stride];
            if (((u & hmask) == prefix) && (((u >> bit) & 1u) == 0u)) cnt0++;
        }
        if (k < cnt0) { /* bit stays 0 */ }
        else { prefix |= 1u << bit; k -= cnt0; }
    }
    return prefix;
}

__global__ __launch_bounds__(128) void ap_colthr(const float* __restrict__ blur,
                                                 float* __restrict__ thrCol) {
    __shared__ unsigned tile[128 * 128];   // 64 KB, conflict-free col reads
    int tid = threadIdx.x;
    int col = blockIdx.x * 128 + tid;
    for (int r = 0; r < 128; ++r) {
        unsigned u = 0xFFFFFFFFu;
        if (col < NFRAMES) u = f2o(blur[(size_t)r * TW + col]);
        tile[r * 128 + tid] = u;
    }
    __syncthreads();
    unsigned p95 = ap_radix_select(&tile[tid], 128, 128, 95);
    unsigned p96 = ap_radix_select(&tile[tid], 128, 128, 96);
    if (col < NFRAMES) {
        float v95 = o2f(p95), v96 = o2f(p96);
        thrCol[col] = v95 + 0.25f * (v96 - v95);
    }
}

// ---------------- K6: final mask ------------------------------------------
__global__ __launch_bounds__(256) void ap_mask(const float* __restrict__ blur,
                                               const float* __restrict__ thrRow,
                                               const float* __restrict__ thrCol,
                                               float* __restrict__ out) {
    int t = blockIdx.x * 256 + threadIdx.x;
    int m = blockIdx.y;
    if (t >= NFRAMES) return;
    float b = blur[(size_t)m * TW + t];
    out[(size_t)m * NFRAMES + t] = (b > thrRow[m] && b > thrCol[t]) ? 1.0f : 0.0f;
}

// ---------------- launch ---------------------------------------------------
extern "C" void kernel_launch(void* const* d_in, const int* in_sizes, int n_in,
                              void* d_out, int out_size, void* d_ws, size_t ws_size,
                              hipStream_t stream) {
    (void)in_sizes; (void)n_in; (void)out_size; (void)ws_size;
    const float* wave = (const float*)d_in[0];
    float* out = (float*)d_out;
    char* ws = (char*)d_ws;
    // workspace layout (bytes, 256-aligned)
    float* ccosT  = (float*)(ws + 0);          // 528*1024*4 = 2,162,688
    float* csinT  = (float*)(ws + 2162688);    // 2,162,688
    float* fbT    = (float*)(ws + 4325376);    // 128*528*4  =   270,336
    float* spec   = (float*)(ws + 4595712);    // 27008*528*4 = 57,040,896
    float* mel    = (float*)(ws + 61636608);   // 128*27008*4 = 13,828,096
    float* blur   = (float*)(ws + 75464704);   // 13,828,096
    float* thrRow = (float*)(ws + 89292800);   // 512
    float* thrCol = (float*)(ws + 89293312);   // 27008*4

    ap_init_dft<<<(KFP * NFFT + 255) / 256, 256, 0, stream>>>(ccosT, csinT);
    ap_init_fb <<<(NMELS * KFP + 255) / 256, 256, 0, stream>>>(fbT);
    ap_spec    <<<dim3(MTILES, 5), 256, 0, stream>>>(wave, ccosT, csinT, spec);
    ap_mel     <<<MTILES, 256, 0, stream>>>(spec, fbT, mel);
    ap_blur    <<<dim3((NFRAMES + 255) / 256, NMELS), 256, 0, stream>>>(mel, blur);
    ap_rowthr  <<<NMELS, 256, 0, stream>>>(blur, thrRow);
    ap_colthr  <<<(NFRAMES + 127) / 128, 128, 0, stream>>>(blur, thrCol);
    ap_mask    <<<dim3((NFRAMES + 255) / 256, NMELS), 256, 0, stream>>>(blur, thrRow, thrCol, out);
}